// SSIM_23124103922415
// MI455X (gfx1250) — compile-verified
//
#include <hip/hip_runtime.h>
#include <math.h>

// ---------------------------------------------------------------------------
// SSIM distance (1 - SSIM style D = sqrt(2 - S)) for [16,3,512,512] fp32.
// Separable 11-tap Gaussian implemented as band-matrix GEMMs on the CDNA5
// fp32 matrix pipe: V_WMMA_F32_16X16X4_F32 (keeps full fp32 accuracy).
// ---------------------------------------------------------------------------

typedef __attribute__((ext_vector_type(2))) float v2f;
typedef __attribute__((ext_vector_type(8))) float v8f;

#define IMG       512
#define OUTD      502          // 512 - 11 + 1
#define NIMG      48           // 16 * 3
#define BT_ROWS   32           // output rows per block
#define BT_COLS   64           // output cols per block
#define GBX       8            // 8 * 64  = 512 >= 502
#define GBY       16           // 16 * 32 = 512 >= 502
#define NBLK      (NIMG * GBY * GBX)
#define IN_ROWS   48           // 32 out rows + 16 (covers 26-row window of top wave)
#define IN_COLS   80           // 64 out cols + 16 (covers 28-col K window)
#define TSTRIDE   81           // LDS row stride (odd vs 64 banks)
#define TMPSTRIDE 17           // tmp row stride

// normalized gauss(11, sigma=1.5)
__constant__ float cG11[11] = {
    0.0010284f, 0.0075988f, 0.0360008f, 0.1093607f, 0.2130056f,
    0.2660117f,
    0.2130056f, 0.1093607f, 0.0360008f, 0.0075988f, 0.0010284f};

__global__ __launch_bounds__(256) void ssim_tiles(const float* __restrict__ X,
                                                  const float* __restrict__ Y,
                                                  float* __restrict__ partial)
{
    __shared__ float s_gtab[48];                       // band table g[i-16], zero outside
    __shared__ float sX[IN_ROWS * TSTRIDE];
    __shared__ float sY[IN_ROWS * TSTRIDE];
    __shared__ float sTmp[8 * 32 * TMPSTRIDE];         // per-wave private tmp
    __shared__ float sRed[256];

    const int tid = threadIdx.x;
    const int img = blockIdx.z;
    const int br  = blockIdx.y * BT_ROWS;              // output-row origin of block
    const int bc  = blockIdx.x * BT_COLS;              // output-col origin of block

    if (tid < 48) {
        int j = tid - 16;
        s_gtab[tid] = (j >= 0 && j < 11) ? cG11[j] : 0.0f;
    }

    // Cooperative tile load (zero-fill outside image; halo beyond valid range
    // only ever multiplies zero band coefficients, but must not be NaN).
    const size_t ibase = (size_t)img * IMG * IMG;
    for (int idx = tid; idx < IN_ROWS * IN_COLS; idx += 256) {
        int r = idx / IN_COLS, c = idx % IN_COLS;
        int gr = br + r, gc = bc + c;
        bool ok = (gr < IMG) && (gc < IMG);
        size_t off = ibase + (size_t)gr * IMG + gc;
        sX[r * TSTRIDE + c] = ok ? X[off] : 0.0f;
        sY[r * TSTRIDE + c] = ok ? Y[off] : 0.0f;
    }
    __syncthreads();

    const int lane = tid & 31;
    const int wave = tid >> 5;
    const int n    = lane & 15;        // N (or M) index of this lane
    const int h    = lane >> 4;        // half-wave -> K offset of 2
    const int wr   = (wave >> 2) * 16; // wave subtile row within block tile
    const int wc   = (wave & 3) * 16;  // wave subtile col within block tile

    // Gaussian band fragments: identical register image serves as
    // B(4x16) for the horizontal pass and A(16x4) for the vertical pass,
    // since fp32 A/B layouts both map K = vgpr + 2*h, M/N = lane&15.
    v2f gf[7];
#pragma unroll
    for (int c = 0; c < 7; ++c) {
        int q0 = 4 * c + 2 * h;
        v2f t;
        t.x = s_gtab[q0 - n + 16];
        t.y = s_gtab[q0 + 1 - n + 16];
        gf[c] = t;
    }

    float* tmp = &sTmp[wave * 32 * TMPSTRIDE];
    v8f res[5];

#pragma unroll
    for (int f = 0; f < 5; ++f) {
        // ---- horizontal pass: two 16-row tmp tiles (rows wr .. wr+31) ----
#pragma unroll
        for (int rt = 0; rt < 2; ++rt) {
            v8f acc = {};
            const int rowA = wr + rt * 16 + n;              // A-matrix row (M = lane&15)
            const float* rowX = &sX[rowA * TSTRIDE];
            const float* rowY = &sY[rowA * TSTRIDE];
#pragma unroll
            for (int c = 0; c < 7; ++c) {
                int col = wc + 4 * c + 2 * h;
                v2f a;
                if (f == 0)      { a.x = rowX[col];             a.y = rowX[col + 1]; }
                else if (f == 1) { a.x = rowY[col];             a.y = rowY[col + 1]; }
                else if (f == 2) { float x0 = rowX[col], x1 = rowX[col + 1];
                                   a.x = x0 * x0;               a.y = x1 * x1; }
                else if (f == 3) { float y0 = rowY[col], y1 = rowY[col + 1];
                                   a.x = y0 * y0;               a.y = y1 * y1; }
                else             { a.x = rowX[col] * rowY[col];
                                   a.y = rowX[col + 1] * rowY[col + 1]; }
                acc = __builtin_amdgcn_wmma_f32_16x16x4_f32(
                    false, a, false, gf[c], (short)0, acc, false, false);
            }
            // D layout: vgpr v -> row (v + 8*h), col n
#pragma unroll
            for (int v = 0; v < 8; ++v)
                tmp[(rt * 16 + v + 8 * h) * TMPSTRIDE + n] = acc[v];
        }

        // ---- vertical pass: out(16x16) = Gv(16x28) x tmp(28x16) ----
        v8f acc = {};
#pragma unroll
        for (int c = 0; c < 7; ++c) {
            int q = 4 * c + 2 * h;
            v2f b;
            b.x = tmp[q * TMPSTRIDE + n];
            b.y = tmp[(q + 1) * TMPSTRIDE + n];
            acc = __builtin_amdgcn_wmma_f32_16x16x4_f32(
                false, gf[c], false, b, (short)0, acc, false, false);
        }
        res[f] = acc;
    }

    // ---- SSIM map + masked local sum (8 output pixels per lane) ----
    const float C1 = 1e-4f, C2 = 9e-4f;
    float lsum = 0.0f;
#pragma unroll
    for (int v = 0; v < 8; ++v) {
        int orow = br + wr + v + 8 * h;
        int ocol = bc + wc + n;
        float mu1 = res[0][v], mu2 = res[1][v];
        float xx  = res[2][v], yy  = res[3][v], xy = res[4][v];
        float mu1s = mu1 * mu1, mu2s = mu2 * mu2, mu12 = mu1 * mu2;
        float s1 = xx - mu1s, s2 = yy - mu2s, s12 = xy - mu12;
        float S1 = (2.0f * mu12 + C1) / (mu1s + mu2s + C1);
        float S2 = (2.0f * s12 + C2) / (s1 + s2 + C2);
        float S  = fminf(S1 + S2, 2.0f);
        float d  = sqrtf(fmaxf(2.0f - S, 0.0f));
        if (orow < OUTD && ocol < OUTD) lsum += d;
    }

    // deterministic block reduction, one partial per block
    sRed[tid] = lsum;
    __syncthreads();
#pragma unroll
    for (int s = 128; s > 0; s >>= 1) {
        if (tid < s) sRed[tid] += sRed[tid + s];
        __syncthreads();
    }
    if (tid == 0)
        partial[((size_t)blockIdx.z * GBY + blockIdx.y) * GBX + blockIdx.x] = sRed[0];
}

__global__ __launch_bounds__(256) void reduce_partials(const float* __restrict__ partial,
                                                       float* __restrict__ out)
{
    __shared__ float sRed[256];
    int tid = threadIdx.x;
    float s = 0.0f;
    for (int i = tid; i < NBLK; i += 256) s += partial[i];   // fixed order -> deterministic
    sRed[tid] = s;
    __syncthreads();
#pragma unroll
    for (int k = 128; k > 0; k >>= 1) {
        if (tid < k) sRed[tid] += sRed[tid + k];
        __syncthreads();
    }
    if (tid == 0)
        out[0] = sRed[0] * (1.0f / ((float)NIMG * (float)OUTD * (float)OUTD));
}

extern "C" void kernel_launch(void* const* d_in, const int* in_sizes, int n_in,
                              void* d_out, int out_size, void* d_ws, size_t ws_size,
                              hipStream_t stream)
{
    const float* X = (const float*)d_in[0];
    const float* Y = (const float*)d_in[1];
    float* partial = (float*)d_ws;     // NBLK floats = 24 KB scratch

    dim3 grid(GBX, GBY, NIMG);
    ssim_tiles<<<grid, 256, 0, stream>>>(X, Y, partial);
    reduce_partials<<<1, 256, 0, stream>>>(partial, (float*)d_out);
}